// ExpertChoiceRouter_28080496181688
// MI455X (gfx1250) — compile-verified
//
#include <hip/hip_runtime.h>
#include <hip/hip_bf16.h>
#include <math.h>

#define T_TOKENS 32768
#define DM       1024
#define N_EXP    64
#define CAP      1024
#define K2_THREADS 1024
#define CHUNK (T_TOKENS / K2_THREADS)   // 32 tokens per thread

typedef __attribute__((ext_vector_type(2))) float v2f;
typedef __attribute__((ext_vector_type(8))) float v8f;

// ---------------------------------------------------------------------------
// Kernel 1: logits = x @ W + b using V_WMMA_F32_16X16X4_F32 (exact fp32 math).
// Block = 256 threads = 8 waves (wave32). Each wave: 16-row M tile, N=64 via
// four 16x16 f32 accumulators. K stepped by 4 per WMMA.
// A (16x4 f32) lane layout: lanes 0-15 -> rows, VGPR0=K0/VGPR1=K1;
// lanes 16-31 -> same rows, K2/K3. B (4x16) mirrored: lane&15 = N column.
// ---------------------------------------------------------------------------
__global__ __launch_bounds__(256) void ecr_gemm_wmma(
    const float* __restrict__ x, const float* __restrict__ W,
    const float* __restrict__ bias, float* __restrict__ logits)
{
    const int lane = threadIdx.x & 31;
    const int wave = threadIdx.x >> 5;
    const int half = lane >> 4;          // 0: K pair {0,1}, 1: K pair {2,3}
    const int l    = lane & 15;
    const int row_base = blockIdx.x * 128 + wave * 16;

    v8f c0 = {}, c1 = {}, c2 = {}, c3 = {};
    const float* __restrict__ xrow = x + (size_t)(row_base + l) * DM;

    for (int k = 0; k < DM; k += 4) {
        const int kk = k + 2 * half;
        v2f a;
        a.x = xrow[kk];
        a.y = xrow[kk + 1];

        const float* __restrict__ w0 = W + (size_t)kk * N_EXP + l;
        const float* __restrict__ w1 = w0 + N_EXP;
        v2f b0, b1, b2, b3;
        b0.x = w0[0];  b0.y = w1[0];
        b1.x = w0[16]; b1.y = w1[16];
        b2.x = w0[32]; b2.y = w1[32];
        b3.x = w0[48]; b3.y = w1[48];

        c0 = __builtin_amdgcn_wmma_f32_16x16x4_f32(false, a, false, b0, (short)0, c0, false, false);
        c1 = __builtin_amdgcn_wmma_f32_16x16x4_f32(false, a, false, b1, (short)0, c1, false, false);
        c2 = __builtin_amdgcn_wmma_f32_16x16x4_f32(false, a, false, b2, (short)0, c2, false, false);
        c3 = __builtin_amdgcn_wmma_f32_16x16x4_f32(false, a, false, b3, (short)0, c3, false, false);
    }

    // C/D layout: VGPR j holds row M = j + 8*half, col N = lane&15.
    const int orow = row_base + 8 * half;
    #pragma unroll
    for (int j = 0; j < 8; ++j) {
        const size_t r = (size_t)(orow + j) * N_EXP;
        logits[r + l]      = c0[j] + bias[l];
        logits[r + 16 + l] = c1[j] + bias[16 + l];
        logits[r + 32 + l] = c2[j] + bias[32 + l];
        logits[r + 48 + l] = c3[j] + bias[48 + l];
    }
}

// ---------------------------------------------------------------------------
// Order-preserving float -> uint key (monotonic), and inverse.
// ---------------------------------------------------------------------------
__device__ __forceinline__ unsigned f2key(float f) {
    unsigned u = __float_as_uint(f);
    return (u & 0x80000000u) ? ~u : (u | 0x80000000u);
}
__device__ __forceinline__ float key2f(unsigned u) {
    u = (u & 0x80000000u) ? (u & 0x7FFFFFFFu) : ~u;
    return __uint_as_float(u);
}

// ---------------------------------------------------------------------------
// Kernel 2: per-expert token-axis softmax + exact sorted top-CAP.
// One 1024-thread block per expert. Column is L2-resident (logits = 8MB).
// ---------------------------------------------------------------------------
__global__ __launch_bounds__(K2_THREADS) void ecr_softmax_topk(
    const float* __restrict__ logits, float* __restrict__ probs,
    float* __restrict__ eprobs, float* __restrict__ eidx)
{
    __shared__ float redf[K2_THREADS];
    __shared__ unsigned hist[256];
    __shared__ unsigned scan[K2_THREADS];
    __shared__ unsigned long long cand[CAP];
    __shared__ unsigned s_prefix, s_need;

    const int e   = blockIdx.x;
    const int tid = threadIdx.x;

    // ---- 1) column max ----
    float m = -INFINITY;
    for (int i = tid; i < T_TOKENS; i += K2_THREADS)
        m = fmaxf(m, logits[(size_t)i * N_EXP + e]);
    redf[tid] = m;
    __syncthreads();
    for (int s = K2_THREADS / 2; s > 0; s >>= 1) {
        if (tid < s) redf[tid] = fmaxf(redf[tid], redf[tid + s]);
        __syncthreads();
    }
    const float maxv = redf[0];
    __syncthreads();

    // ---- 2) sum of exp ----
    float su = 0.0f;
    for (int i = tid; i < T_TOKENS; i += K2_THREADS)
        su += expf(logits[(size_t)i * N_EXP + e] - maxv);
    redf[tid] = su;
    __syncthreads();
    for (int s = K2_THREADS / 2; s > 0; s >>= 1) {
        if (tid < s) redf[tid] += redf[tid + s];
        __syncthreads();
    }
    const float invsum = 1.0f / redf[0];
    __syncthreads();

    // ---- 3) write full probs column ----
    for (int i = tid; i < T_TOKENS; i += K2_THREADS)
        probs[(size_t)i * N_EXP + e] =
            expf(logits[(size_t)i * N_EXP + e] - maxv) * invsum;

    // ---- 4) radix select: exact bit pattern of the CAP-th largest key ----
    unsigned prefix = 0;
    unsigned kneed  = CAP;
    for (int byte = 3; byte >= 0; --byte) {
        const unsigned known = (byte == 3) ? 0u : (0xFFFFFFFFu << ((byte + 1) * 8));
        if (tid < 256) hist[tid] = 0;
        __syncthreads();
        for (int i = tid; i < T_TOKENS; i += K2_THREADS) {
            const unsigned u = f2key(logits[(size_t)i * N_EXP + e]);
            if ((u & known) == (prefix & known))
                atomicAdd(&hist[(u >> (byte * 8)) & 255u], 1u);
        }
        __syncthreads();
        if (tid == 0) {
            unsigned cum = 0;
            int b = 255;
            for (; b > 0; --b) {
                const unsigned c = hist[b];
                if (cum + c >= kneed) break;
                cum += c;
            }
            s_prefix = prefix | ((unsigned)b << (byte * 8));
            s_need   = kneed - cum;
        }
        __syncthreads();
        prefix = s_prefix;
        kneed  = s_need;
        __syncthreads();
    }
    const unsigned Tkey = prefix;          // exact key of CAP-th largest
    const unsigned need = kneed;           // how many ==Tkey to keep
    const unsigned cntG = CAP - need;      // strictly-greater count

    // ---- 5) deterministic compaction (per-thread contiguous chunks so that
    //          ties take lowest token indices first, matching top_k) ----
    const int base = tid * CHUNK;
    unsigned g = 0, eq = 0;
    for (int i = 0; i < CHUNK; ++i) {
        const unsigned u = f2key(logits[(size_t)(base + i) * N_EXP + e]);
        if (u > Tkey) ++g; else if (u == Tkey) ++eq;
    }
    // inclusive scan of g
    scan[tid] = g;
    __syncthreads();
    for (int off = 1; off < K2_THREADS; off <<= 1) {
        const unsigned v = (tid >= off) ? scan[tid - off] : 0u;
        __syncthreads();
        scan[tid] += v;
        __syncthreads();
    }
    const unsigned gOff = scan[tid] - g;
    __syncthreads();
    // inclusive scan of eq
    scan[tid] = eq;
    __syncthreads();
    for (int off = 1; off < K2_THREADS; off <<= 1) {
        const unsigned v = (tid >= off) ? scan[tid - off] : 0u;
        __syncthreads();
        scan[tid] += v;
        __syncthreads();
    }
    const unsigned eOff = scan[tid] - eq;
    __syncthreads();

    unsigned gp = gOff;
    unsigned ep = cntG + eOff;
    for (int i = 0; i < CHUNK; ++i) {
        const int tok = base + i;
        const unsigned u = f2key(logits[(size_t)tok * N_EXP + e]);
        if (u > Tkey) {
            cand[gp++] = ((unsigned long long)u << 32) | (unsigned)(~tok);
        } else if (u == Tkey) {
            if (ep < CAP)
                cand[ep] = ((unsigned long long)u << 32) | (unsigned)(~tok);
            ++ep;
        }
    }
    __syncthreads();

    // ---- 6) bitonic sort CAP packed keys, descending
    //         (value desc; ~token makes ties index-ascending) ----
    for (unsigned kk = 2; kk <= CAP; kk <<= 1) {
        for (unsigned j = kk >> 1; j > 0; j >>= 1) {
            const unsigned ixj = (unsigned)tid ^ j;
            if (ixj > (unsigned)tid) {
                const unsigned long long a = cand[tid];
                const unsigned long long b = cand[ixj];
                const bool doswap = (((unsigned)tid & kk) == 0) ? (a < b) : (a > b);
                if (doswap) { cand[tid] = b; cand[ixj] = a; }
            }
            __syncthreads();
        }
    }

    // ---- 7) emit sorted top-CAP probs + indices (as float) ----
    const unsigned long long ck = cand[tid];
    const unsigned u   = (unsigned)(ck >> 32);
    const unsigned tok = ~((unsigned)ck);
    const float p = expf(key2f(u) - maxv) * invsum;
    eprobs[(size_t)e * CAP + tid] = p;
    eidx  [(size_t)e * CAP + tid] = (float)tok;
}

// ---------------------------------------------------------------------------
// Host launch: out = [logits | probs | expert_probs | expert_indices] (float)
// ---------------------------------------------------------------------------
extern "C" void kernel_launch(void* const* d_in, const int* in_sizes, int n_in,
                              void* d_out, int out_size, void* d_ws, size_t ws_size,
                              hipStream_t stream) {
    const float* x    = (const float*)d_in[0];
    const float* W    = (const float*)d_in[1];
    const float* bias = (const float*)d_in[2];
    // d_in[3] = k (unused by reference math), d_in[4] = expert_capacity (=1024)

    float* out    = (float*)d_out;
    float* logits = out;
    float* probs  = out + (size_t)T_TOKENS * N_EXP;
    float* eprobs = out + 2 * (size_t)T_TOKENS * N_EXP;
    float* eidx   = eprobs + (size_t)N_EXP * CAP;

    ecr_gemm_wmma<<<T_TOKENS / 128, 256, 0, stream>>>(x, W, bias, logits);
    ecr_softmax_topk<<<N_EXP, K2_THREADS, 0, stream>>>(logits, probs, eprobs, eidx);
}